// CenterLoss_46222438039914
// MI455X (gfx1250) — compile-verified
//
#include <hip/hip_runtime.h>

typedef __attribute__((ext_vector_type(2))) float v2f;
typedef __attribute__((ext_vector_type(8))) float v8f;

// Problem shape (fixed by the reference)
constexpr int N = 16, D = 64, C = 80, WH = 16384;

// Tiling
constexpr int W_CHUNK = 512;              // w-span per workgroup
constexpr int W_TILE  = 64;               // w-span per LDS stage
constexpr int TILES   = W_CHUNK / W_TILE; // 8
constexpr int LDS_STRIDE = W_TILE + 4;    // 68: keeps float4 16B alignment, avoids bank conflicts
constexpr int NTHREADS = 320;             // 10 wave32 waves; 20 WMMA tiles -> 2 per wave
constexpr int NWG = (WH / W_CHUNK) * N;   // 512 workgroups

// ---- async global -> LDS path (CDNA5 GLOBAL_LOAD_ASYNC_TO_LDS_B128) ------
#if defined(__AMDGCN__) && __has_builtin(__builtin_amdgcn_global_load_async_to_lds_b128)
#define ASYNC_LDS 1
#else
#define ASYNC_LDS 0
#endif

#if ASYNC_LDS
// Param types per hipcc diagnostic: arg0 = addrspace(1) ("__device__") pointer
// to 'int __attribute__((vector_size(16)))', non-const. arg1 = LDS-side twin.
typedef int b128_t __attribute__((__vector_size__(4 * sizeof(int))));
typedef __attribute__((address_space(1))) b128_t* g128_t;
typedef __attribute__((address_space(3))) b128_t* l128_t;
#endif

__device__ __forceinline__ void copy16(const float* __restrict__ g, float* l) {
#if ASYNC_LDS
  // per-lane 16B async transfer, tracked by ASYNCcnt (no VGPR data staging)
  __builtin_amdgcn_global_load_async_to_lds_b128(
      (g128_t)(unsigned long long)g,
      (l128_t)(unsigned int)(unsigned long long)l,  // low 32 bits of flat = LDS offset
      0, 0);
#else
  *(float4*)l = *(const float4*)g;
#endif
}

__device__ __forceinline__ void wait_async_loads() {
#if ASYNC_LDS
#if __has_builtin(__builtin_amdgcn_s_wait_asynccnt)
  __builtin_amdgcn_s_wait_asynccnt(0);
#else
  asm volatile("s_wait_asynccnt 0" ::: "memory");
#endif
#endif
}

__global__ __launch_bounds__(NTHREADS)
void center_loss_main(const float* __restrict__ x, const float* __restrict__ c,
                      const float* __restrict__ act, float* __restrict__ partials) {
  __shared__ float x_lds[2][D * LDS_STRIDE];   // double-buffered 64 x 68 floats
  __shared__ float a_lds[2][C * LDS_STRIDE];   // double-buffered 80 x 68 floats
  __shared__ float red[512];

  const int tid  = threadIdx.x;
  const int lane = tid & 31;
  const int wave = tid >> 5;
  const int n     = blockIdx.y;
  const int wbase = blockIdx.x * W_CHUNK;

  const float* xn = x   + (size_t)n * D * WH;
  const float* an = act + (size_t)n * C * WH;

  // ---- persistent per-thread scalar state -------------------------------
  float s1    = 0.f;   // threads 0..63 (w owners): sum_k x2[k]*A[k]
  float b_own = 0.f;   // threads 64..143 (cc owners): partial B[cc]
  float c2_own = 0.f;  // ||c[:,cc]||^2 for the cc owner
  if (tid >= 64 && tid < 64 + C) {
    const int cc = tid - 64;
    for (int d = 0; d < D; ++d) { float v = c[d * C + cc]; c2_own += v * v; }
  }

  // ---- WMMA tile assignment: t = wave*2, wave*2+1 over 4(d) x 5(cc) -----
  // Consecutive pair always shares the cc-tile -> shared B fragment.
  const int t0  = wave * 2;
  const int db0 = (t0 & 3) * 16;
  const int db1 = ((t0 + 1) & 3) * 16;
  const int cb  = (t0 >> 2) * 16;
  const int rlo = lane & 15;          // M (A) / N (B,C,D) index within the 16-tile
  const int whi = (lane >> 4) * 2;    // K offset for upper half-wave (ISA A/B layout)
  const int aoff0 = (db0 + rlo) * LDS_STRIDE + whi;
  const int aoff1 = (db1 + rlo) * LDS_STRIDE + whi;
  const int boff  = (cb  + rlo) * LDS_STRIDE + whi;

  v8f acc0 = {}; v8f acc1 = {};

  // ---- prologue: kick off tile 0 into buffer 0 --------------------------
  {
    const int w0 = wbase;
    for (int i = tid; i < (D * W_TILE) / 4; i += NTHREADS) {
      const int row = i >> 4, col = (i & 15) * 4;
      copy16(xn + (size_t)row * WH + w0 + col, &x_lds[0][row * LDS_STRIDE + col]);
    }
    for (int i = tid; i < (C * W_TILE) / 4; i += NTHREADS) {
      const int row = i >> 4, col = (i & 15) * 4;
      copy16(an + (size_t)row * WH + w0 + col, &a_lds[0][row * LDS_STRIDE + col]);
    }
  }

  for (int wt = 0; wt < TILES; ++wt) {
    const int cur = wt & 1;

    // own async loads done, then cross-wave visibility
    wait_async_loads();
    __syncthreads();

    // ---- issue next tile's loads into the other buffer (overlapped) -----
    if (wt + 1 < TILES) {
      const int w1 = wbase + (wt + 1) * W_TILE;
      float* xb = x_lds[1 - cur];
      float* ab = a_lds[1 - cur];
      for (int i = tid; i < (D * W_TILE) / 4; i += NTHREADS) {
        const int row = i >> 4, col = (i & 15) * 4;
        copy16(xn + (size_t)row * WH + w1 + col, &xb[row * LDS_STRIDE + col]);
      }
      for (int i = tid; i < (C * W_TILE) / 4; i += NTHREADS) {
        const int row = i >> 4, col = (i & 15) * 4;
        copy16(an + (size_t)row * WH + w1 + col, &ab[row * LDS_STRIDE + col]);
      }
    }

    const float* xcur = x_lds[cur];
    const float* acur = a_lds[cur];

    // ---- side sums from the LDS tile (conflict-free column reads) -------
    if (tid < W_TILE) {            // this thread owns w = tid
      float A = 0.f, x2 = 0.f;
      for (int cc = 0; cc < C; ++cc) A += acur[cc * LDS_STRIDE + tid];
      for (int d = 0; d < D; ++d) { float v = xcur[d * LDS_STRIDE + tid]; x2 += v * v; }
      s1 += x2 * A;
    } else if (tid < 64 + C) {     // this thread owns cc = tid-64
      const int cc = tid - 64;
      float s = 0.f;
      for (int w = 0; w < W_TILE; ++w) s += acur[cc * LDS_STRIDE + w];
      b_own += s;
    }

    // ---- WMMA phase (full EXEC): M[d,cc] += x_tile * act_tile^T ---------
    #pragma unroll
    for (int kk = 0; kk < W_TILE / 4; ++kk) {
      v2f b  = *(const v2f*)&acur[boff  + kk * 4];   // B frag shared by both tiles
      v2f a0 = *(const v2f*)&xcur[aoff0 + kk * 4];
      v2f a1 = *(const v2f*)&xcur[aoff1 + kk * 4];
      acc0 = __builtin_amdgcn_wmma_f32_16x16x4_f32(false, a0, false, b, (short)0, acc0, false, false);
      acc1 = __builtin_amdgcn_wmma_f32_16x16x4_f32(false, a1, false, b, (short)0, acc1, false, false);
    }
  }

  // ---- fold everything into one per-thread partial ----------------------
  float p = s1;
  if (tid >= 64 && tid < 64 + C) p += b_own * c2_own;   // c2 . B term
  const int mrow = (lane >> 4) * 8;                     // C/D layout: VGPR r -> row r / r+8
  #pragma unroll
  for (int r = 0; r < 8; ++r) {
    p -= 2.f * acc0[r] * c[(db0 + r + mrow) * C + (cb + rlo)];
    p -= 2.f * acc1[r] * c[(db1 + r + mrow) * C + (cb + rlo)];
  }

  // ---- deterministic block reduction -> one partial per WG --------------
  __syncthreads();   // everyone done with LDS tiles; reuse red[]
  red[tid] = p;
  if (tid < 512 - NTHREADS) red[NTHREADS + tid] = 0.f;
  __syncthreads();
  for (int off = 256; off > 0; off >>= 1) {
    if (tid < off) red[tid] += red[tid + off];
    __syncthreads();
  }
  if (tid == 0) partials[blockIdx.y * gridDim.x + blockIdx.x] = red[0];
}

__global__ __launch_bounds__(256)
void center_loss_finalize(const float* __restrict__ partials, float* __restrict__ out, int nparts) {
  __shared__ float red[256];
  const int tid = threadIdx.x;
  float s = 0.f;
  for (int i = tid; i < nparts; i += 256) s += partials[i];
  red[tid] = s;
  __syncthreads();
  for (int off = 128; off > 0; off >>= 1) {
    if (tid < off) red[tid] += red[tid + off];
    __syncthreads();
  }
  if (tid == 0) out[0] = fmaxf(red[0], 1e-6f);
}

extern "C" void kernel_launch(void* const* d_in, const int* in_sizes, int n_in,
                              void* d_out, int out_size, void* d_ws, size_t ws_size,
                              hipStream_t stream) {
  (void)in_sizes; (void)n_in; (void)out_size; (void)ws_size;
  const float* x   = (const float*)d_in[0];   // [N, D, WH]
  const float* c   = (const float*)d_in[1];   // [D, C]
  const float* act = (const float*)d_in[2];   // [N, C, WH]
  float* out      = (float*)d_out;            // scalar
  float* partials = (float*)d_ws;             // NWG floats of scratch

  dim3 grid(WH / W_CHUNK, N);                 // (32, 16) = 512 WGs
  center_loss_main<<<grid, NTHREADS, 0, stream>>>(x, c, act, partials);
  center_loss_finalize<<<1, 256, 0, stream>>>(partials, out, NWG);
}